// SubbandFilter_2293512536585
// MI455X (gfx1250) — compile-verified
//
#include <hip/hip_runtime.h>

// ---------------------------------------------------------------------------
// Problem constants (from the reference)
// ---------------------------------------------------------------------------
#define SR_I      44100
#define L_IN      441000
#define FACTOR    10
#define FLEN      10001
#define UP_TAPS   201
#define NBANDS    6
#define NSIG      4                      // B*C
#define NPOS      (L_IN / FACTOR)        // 44100 decimated positions / signal
#define PTOT      (NSIG * NPOS)          // 176400
#define TPS       ((NPOS + 15) / 16)     // 2757 tiles per signal
#define NTILES    (NSIG * TPS)           // 11028
#define PI_F      3.14159265358979323846f
#define TWO_PI_D  6.2831853071795864769

// K-dimension chunking for the TDM-staged WMMA GEMM
#define KC        256                    // taps per LDS chunk
#define NCHUNK    40
#define KPAD2     (KC * NCHUNK)          // 10240 (FLEN zero-padded)
#define SFSTRIDE  (KC + 4)               // LDS row stride for F (bank-conflict pad)
#define HALF      ((FLEN - 1) / 2)       // 5000
#define XT        408                    // x window tile length (10*15 + KC, rounded)
#define XW        416                    // LDS stride per wave's x window
#define PADL      451360                 // zero-padded signal length per channel
#define BUFSZ     (16 * SFSTRIDE + 8 * XW)   // one double-buffer slice (floats)

typedef unsigned int       u32;
typedef unsigned long long u64;
typedef __attribute__((ext_vector_type(2))) float v2f;
typedef __attribute__((ext_vector_type(8))) float v8f;
typedef __attribute__((ext_vector_type(4))) u32   v4u;
typedef __attribute__((ext_vector_type(8))) u32   v8u;

// ---------------------------------------------------------------------------
// Tensor Data Mover issue: 2-D tile (tile_d0 x tile_d1) of 4-byte elements
// from global -> LDS.  D# packing per CDNA5 ISA #8.
//   group0: [1:0]=count=1 | [63:32]=lds_addr | [120:64]=global_addr | type=2
//   group1: cfg(data_size/pad) | tensor dims | tile dims | dim0 stride
// Tracked with TENSORcnt (in order per wave); EXEC is ignored.
// ---------------------------------------------------------------------------
static __device__ __forceinline__ void tdm_load_2d(u32 lds_byte, u64 gaddr,
                                                   u32 cfg,
                                                   u32 tensor_d0, u32 tensor_d1,
                                                   u32 tile_d0,   u32 tile_d1,
                                                   u32 stride0) {
    union { u32 a[4]; v4u v; } g0;
    union { u32 a[8]; v8u v; } g1;
    g0.a[0] = 1u;                                            // count=1 (valid)
    g0.a[1] = lds_byte;                                      // lds_addr
    g0.a[2] = (u32)gaddr;                                    // global_addr[31:0]
    g0.a[3] = ((u32)(gaddr >> 32) & 0x01FFFFFFu)             // global_addr[56:32]
              | 0x80000000u;                                 // type=2 ("image")
    g1.a[0] = cfg;                                           // data_size=4B [+pad]
    g1.a[1] = (tensor_d0 & 0xFFFFu) << 16;                   // tensor_dim0[15:0]
    g1.a[2] = (tensor_d0 >> 16) | ((tensor_d1 & 0xFFFFu) << 16);
    g1.a[3] = (tensor_d1 >> 16) | (tile_d0 << 16);           // tile_dim0
    g1.a[4] = tile_d1;                                       // tile_dim1, tile_dim2=0
    g1.a[5] = stride0;                                       // tensor_dim0_stride lo
    g1.a[6] = 0;
    g1.a[7] = 0;
    asm volatile("tensor_load_to_lds %0, %1" :: "s"(g0.v), "s"(g1.v) : "memory");
}

#define TDM_CFG_PLAIN   (2u << 16)                            // 4-byte elements
// pad_enable | pad_interval=7 (every 256 DWORDs) | pad_amount=3 (+4 DWORDs):
// turns packed 16xKC rows into SFSTRIDE-strided LDS rows (no bank conflicts).
#define TDM_CFG_FPAD    ((2u << 16) | (1u << 20) | (7u << 22) | (3u << 25))

// ---------------------------------------------------------------------------
// K0: zero-padded signal: xpad[sig][p] = x[sig][p - HALF] (0 outside).
// Removes every bounds check from the WMMA hot loop.
// ---------------------------------------------------------------------------
__global__ void build_xpad(const float* __restrict__ x, float* __restrict__ xp) {
    int i = blockIdx.x * blockDim.x + threadIdx.x;
    if (i >= NSIG * PADL) return;
    int sig = i / PADL;
    int p   = i % PADL;
    int t   = p - HALF;
    xp[i] = (t >= 0 && t < L_IN) ? x[(size_t)sig * L_IN + t] : 0.0f;
}

// ---------------------------------------------------------------------------
// K1a: rows 0..5 of F = time-reversed band filters; rows 12..15 + K-pad zero.
// F is 16 x KPAD2 row-major.
// ---------------------------------------------------------------------------
__global__ void build_fmat_base(const float* __restrict__ w,
                                float* __restrict__ F) {
    int i = blockIdx.x * blockDim.x + threadIdx.x;
    if (i >= 16 * KPAD2) return;
    int row = i / KPAD2;
    int j   = i % KPAD2;
    float v = 0.0f;
    if (row < NBANDS && j < FLEN) v = w[row * FLEN + (FLEN - 1 - j)];
    F[i] = v;
}

// ---------------------------------------------------------------------------
// K1b: rows 6..11 of F = time-reversed H(w_k); ideal Hilbert kernel 2/(pi*m),
// odd m.  conv(analytic(x), w) == conv(x, w) + i*conv(x, H(w)).
// ---------------------------------------------------------------------------
__global__ void build_fmat_hilbert(const float* __restrict__ w,
                                   float* __restrict__ F) {
    int i = blockIdx.x * blockDim.x + threadIdx.x;
    if (i >= NBANDS * KPAD2) return;
    int k = i / KPAD2;
    int j = i % KPAD2;
    float h = 0.0f;
    if (j < FLEN) {
        int n = (FLEN - 1) - j;
        const float* wk = w + k * FLEN;
        const float two_over_pi = 0.63661977236758134f;
        for (int m = 1; m < FLEN; m += 2) {
            float coef = two_over_pi / (float)m;
            float a = (n - m >= 0)   ? wk[n - m] : 0.0f;
            float b = (n + m < FLEN) ? wk[n + m] : 0.0f;
            h += coef * (a - b);
        }
    }
    F[(NBANDS + k) * KPAD2 + j] = h;
}

// ---------------------------------------------------------------------------
// K2: double-buffered TDM-staged WMMA GEMM.
//   D[row, pos] = sum_k F[row, k] * xpad[sig, 10*pos + k]
// One wave per 16-position tile, 8 waves/block sharing F chunks in LDS.
// Pipeline: TDM for chunk c+1 (buffer b^1) is issued before computing chunk c
// (buffer b); TENSORcnt completes in order per wave, so waiting tensorcnt<=2
// (wave0: F+X prefetched) / <=1 (others: X prefetched) guarantees the current
// chunk landed while the next stays in flight.  WAR on buffer reuse is fenced
// by the barrier at the end of the previous compute phase.
// ---------------------------------------------------------------------------
__global__ void __launch_bounds__(256) wmma_subband(
        const float* __restrict__ xpad,
        const float* __restrict__ F,
        float* __restrict__ esub) {
    __shared__ float smem[2 * BUFSZ];                  // ~60 KB, double buffer

    const int lane = threadIdx.x & 31;
    const int wave = threadIdx.x >> 5;
    int tile = blockIdx.x * 8 + wave;
    const bool active = (tile < NTILES);
    if (!active) tile = 0;                 // tail waves duplicate tile 0 (no store)

    const int sig     = tile / TPS;
    const int m0      = (tile % TPS) * 16;
    const int halfsel = lane >> 4;         // 0: lanes 0-15, 1: lanes 16-31
    const int l16     = lane & 15;

    const float* xw_base = xpad + (size_t)sig * PADL + 10 * m0;

    // Per-buffer LDS byte offsets for the TDM descriptors.
    u32 ldsF[2], ldsX[2];
    ldsF[0] = (u32)(size_t)(smem);
    ldsF[1] = (u32)(size_t)(smem + BUFSZ);
    ldsX[0] = (u32)(size_t)(smem + 16 * SFSTRIDE + wave * XW);
    ldsX[1] = (u32)(size_t)(smem + BUFSZ + 16 * SFSTRIDE + wave * XW);

    v8f acc = {};

    // Prologue: chunk 0 into buffer 0.
    if (wave == 0)
        tdm_load_2d(ldsF[0], (u64)(size_t)(F), TDM_CFG_FPAD,
                    KPAD2, 16, KC, 16, KPAD2);
    tdm_load_2d(ldsX[0], (u64)(size_t)(xw_base), TDM_CFG_PLAIN,
                XT, 1, XT, 1, XT);

    for (int c = 0; c < NCHUNK; ++c) {
        const int b = c & 1;

        // Prefetch chunk c+1 into the other buffer (safe: that buffer was
        // last read during compute(c-1), fenced by the trailing barrier).
        if (c + 1 < NCHUNK) {
            if (wave == 0)
                tdm_load_2d(ldsF[b ^ 1], (u64)(size_t)(F + (c + 1) * KC),
                            TDM_CFG_FPAD, KPAD2, 16, KC, 16, KPAD2);
            tdm_load_2d(ldsX[b ^ 1], (u64)(size_t)(xw_base + (c + 1) * KC),
                        TDM_CFG_PLAIN, XT, 1, XT, 1, XT);
            if (wave == 0) __builtin_amdgcn_s_wait_tensorcnt(2);
            else           __builtin_amdgcn_s_wait_tensorcnt(1);
        } else {
            __builtin_amdgcn_s_wait_tensorcnt(0);
        }
        __syncthreads();                    // publish wave0's F chunk to all

        // A: row = l16, K pair {0,1} or {2,3};  B: col = l16, same K pair.
        const float* fb = smem + b * BUFSZ + l16 * SFSTRIDE + 2 * halfsel;
        const float* xb = smem + b * BUFSZ + 16 * SFSTRIDE + wave * XW
                        + 10 * l16 + 2 * halfsel;
#pragma unroll
        for (int kk = 0; kk < KC; kk += 4) {
            v2f a, bb;
            a.x  = fb[kk];
            a.y  = fb[kk + 1];
            bb.x = xb[kk];
            bb.y = xb[kk + 1];
            acc = __builtin_amdgcn_wmma_f32_16x16x4_f32(
                      false, a, false, bb, (short)0, acc, false, false);
        }
        __syncthreads();                    // everyone done reading buffer b
    }

    const int m = m0 + l16;
    if (active && m < NPOS) {
        const int rbase = halfsel * 8;
        const size_t col = (size_t)sig * NPOS + m;
#pragma unroll
        for (int j = 0; j < 8; ++j) {
            int row = rbase + j;
            if (row < 2 * NBANDS)                           // 12 live rows
                esub[(size_t)row * PTOT + col] = acc[j];
        }
    }
}

// ---------------------------------------------------------------------------
// K3: per-band complex modulation e *= exp(i*omega_k*10*q), phase reduced
// mod 2*pi in double (omega*t reaches ~1.3e6 rad).  In place.
// ---------------------------------------------------------------------------
__global__ void modulate(const float* __restrict__ f_center,
                         float* __restrict__ esub) {
    int i = blockIdx.x * blockDim.x + threadIdx.x;
    if (i >= NBANDS * PTOT) return;
    int k = i / PTOT;
    int r = i % PTOT;
    int q = r % NPOS;

    float omega = -f_center[k] / (SR_I * 0.5f) * PI_F;
    double th   = fmod((double)omega * (double)(FACTOR * q), TWO_PI_D);
    float  s    = __sinf((float)th);
    float  c    = __cosf((float)th);

    float re = esub[(size_t)k * PTOT + r];
    float im = esub[(size_t)(k + NBANDS) * PTOT + r];
    esub[(size_t)k * PTOT + r]            = re * c - im * s;
    esub[(size_t)(k + NBANDS) * PTOT + r] = re * s + im * c;
}

// ---------------------------------------------------------------------------
// K4: zero-stuffed upsample + 201-tap interpolation + band sum.
// Only taps j == t (mod 10) hit nonzero samples: 20-21 taps per output.
// ---------------------------------------------------------------------------
__global__ void upsample_sum(const float* __restrict__ esub,
                             const float* __restrict__ up,
                             float* __restrict__ out) {
    int idx = blockIdx.x * blockDim.x + threadIdx.x;
    if (idx >= NSIG * L_IN) return;
    int sig = idx / L_IN;
    int t   = idx % L_IN;
    int r   = t % FACTOR;
    int q0  = t / FACTOR + FACTOR;

    float accre = 0.0f, accim = 0.0f;
    for (int j = r, q = q0; j < UP_TAPS; j += FACTOR, --q) {
        if (q < 0 || q >= NPOS) continue;
        float uj = up[j];
        size_t base = (size_t)sig * NPOS + q;
#pragma unroll
        for (int k = 0; k < NBANDS; ++k) {
            accre += uj * esub[(size_t)k * PTOT + base];
            accim += uj * esub[(size_t)(k + NBANDS) * PTOT + base];
        }
    }
    out[(size_t)(sig * 2 + 0) * L_IN + t] = accre;
    out[(size_t)(sig * 2 + 1) * L_IN + t] = accim;
}

// ---------------------------------------------------------------------------
// Launch.  d_in = {x, w, f_center, up}; out = (2,2,2,441000) f32.
// Workspace: xpad (NSIG*PADL) | F (16*KPAD2) | esub (12*PTOT)  ~= 16.3 MB.
// Every workspace byte is rewritten before being read each call.
// ---------------------------------------------------------------------------
extern "C" void kernel_launch(void* const* d_in, const int* in_sizes, int n_in,
                              void* d_out, int out_size, void* d_ws, size_t ws_size,
                              hipStream_t stream) {
    (void)in_sizes; (void)n_in; (void)out_size; (void)ws_size;
    const float* x  = (const float*)d_in[0];
    const float* w  = (const float*)d_in[1];
    const float* fc = (const float*)d_in[2];
    const float* up = (const float*)d_in[3];
    float* out  = (float*)d_out;

    float* xpad = (float*)d_ws;
    float* F    = xpad + (size_t)NSIG * PADL;
    float* esub = F + 16 * KPAD2;

    {   // K0: zero-padded signal
        int n = NSIG * PADL;
        build_xpad<<<(n + 255) / 256, 256, 0, stream>>>(x, xpad);
    }
    {   // K1a: reversed band filters + zero padding
        int n = 16 * KPAD2;
        build_fmat_base<<<(n + 255) / 256, 256, 0, stream>>>(w, F);
    }
    {   // K1b: Hilbert rows
        int n = NBANDS * KPAD2;
        build_fmat_hilbert<<<(n + 255) / 256, 256, 0, stream>>>(w, F);
    }
    {   // K2: double-buffered TDM-staged WMMA GEMM (decimation folded in)
        int blocks = (NTILES + 7) / 8;
        wmma_subband<<<blocks, 256, 0, stream>>>(xpad, F, esub);
    }
    {   // K3: complex modulation
        int n = NBANDS * PTOT;
        modulate<<<(n + 255) / 256, 256, 0, stream>>>(fc, esub);
    }
    {   // K4: polyphase upsample + band sum + Re/Im split
        int n = NSIG * L_IN;
        upsample_sum<<<(n + 255) / 256, 256, 0, stream>>>(esub, up, out);
    }
}